// GCN_12395275616828
// MI455X (gfx1250) — compile-verified
//
#include <hip/hip_runtime.h>

// ---------------------------------------------------------------------------
// GCN 3-layer forward for MI455X (gfx1250, wave32).
//   per layer: support = relu?(H) @ W   (WMMA f32 16x16x4, exact fp32)
//              agg[dst] += ew * support[src]  (hw fp32 atomics, L2-resident)
//              agg buffer pre-seeded with bias; ReLU fused into next GEMM.
// All feature dims are compile-time so index math is shifts/masks, not idiv.
// ---------------------------------------------------------------------------

typedef float v2f __attribute__((ext_vector_type(2)));
typedef float v8f __attribute__((ext_vector_type(8)));

#define N_NODES 50000
#define N_EDGES 800000

// ---------------------------------------------------------------------------
// GEMM: Y[N, DOUT] = (RELU_IN ? relu(X) : X)[N, K] @ W[K, DOUT]
// One wave computes one 16x16 output tile; blockDim.x = 32 * ceil(DOUT/16),
// grid.x = N/16 (N must be a multiple of 16; 50000 = 3125*16).
// A-frag (16x4 f32): lane L(<16): {X[m0+L][k], X[m0+L][k+1]}, lane L+16: k+2,k+3
// B-frag (4x16 f32): lane L(<16): {W[k][n0+L], W[k+1][n0+L]},  lane L+16: k+2,k+3
// D (16x16 f32): VGPR r -> row m0 + r + 8*(lane>=16), col n0 + (lane&15)
// ---------------------------------------------------------------------------
template <int K, int DOUT, bool RELU_IN>
__global__ __launch_bounds__(256) void gemm_wmma_f32(
    const float* __restrict__ X, const float* __restrict__ W,
    float* __restrict__ Y, int N) {
  constexpr bool GUARD = (DOUT % 16) != 0;   // only the DOUT=10 layer
  const int wave  = threadIdx.x >> 5;
  const int lane  = threadIdx.x & 31;
  const int n0    = wave * 16;               // column tile base (wave-uniform)
  const int m0    = blockIdx.x * 16;         // row tile base
  const int lhalf = lane >> 4;               // 0: lanes 0-15, 1: lanes 16-31
  const int l16   = lane & 15;

  const int  rowA  = m0 + l16;
  const int  colB  = n0 + l16;
  const bool colOK = !GUARD || (colB < DOUT);
  const float* xrow = X + (size_t)rowA * K;
  const int  kBase = 2 * lhalf;

  v8f acc = {0.f, 0.f, 0.f, 0.f, 0.f, 0.f, 0.f, 0.f};
#pragma unroll 8
  for (int k = 0; k < K; k += 4) {
    float ax = xrow[k + kBase + 0];
    float ay = xrow[k + kBase + 1];
    if (RELU_IN) { ax = fmaxf(ax, 0.f); ay = fmaxf(ay, 0.f); }
    v2f a = {ax, ay};
    v2f b;
    if (GUARD) {
      b.x = colOK ? W[(size_t)(k + kBase + 0) * DOUT + colB] : 0.f;
      b.y = colOK ? W[(size_t)(k + kBase + 1) * DOUT + colB] : 0.f;
    } else {
      b.x = W[(size_t)(k + kBase + 0) * DOUT + colB];
      b.y = W[(size_t)(k + kBase + 1) * DOUT + colB];
    }
    acc = __builtin_amdgcn_wmma_f32_16x16x4_f32(
        /*neg_a=*/false, a, /*neg_b=*/false, b,
        /*c_mod=*/(short)0, acc, /*reuse_a=*/false, /*reuse_b=*/false);
  }

  if (colOK) {
#pragma unroll
    for (int r = 0; r < 8; ++r) {
      Y[(size_t)(m0 + r + 8 * lhalf) * DOUT + colB] = acc[r];
    }
  }
}

// ---------------------------------------------------------------------------
// Seed aggregation buffer with broadcast bias: A[n][f] = b[f]
// ---------------------------------------------------------------------------
template <int D>
__global__ __launch_bounds__(256) void init_bias(
    float* __restrict__ A, const float* __restrict__ b, int N) {
  size_t i = (size_t)blockIdx.x * blockDim.x + threadIdx.x;
  if (i < (size_t)N * D) A[i] = b[(int)(i % D)];
}

// ---------------------------------------------------------------------------
// Edge aggregation, vectorized (D % 4 == 0):
//   A[dst[e]][f..f+3] += ew[e] * S[src[e]][f..f+3]
// Consecutive lanes cover consecutive float4 chunks of one edge row:
// coalesced 16B gathers + hardware fp32 scatter-atomics resolving in L2
// (support and agg buffers are both L2-resident: <= 25.6 MB each vs 192 MB).
// D compile-time -> chunk index math is shift/mask, not integer division.
// ---------------------------------------------------------------------------
template <int D>
__global__ __launch_bounds__(256) void agg_vec4(
    const float* __restrict__ S, const int* __restrict__ src,
    const int* __restrict__ dst, const float* __restrict__ ew,
    float* __restrict__ A, int E) {
  constexpr int CH = D >> 2;                 // float4 chunks per row (pow2)
  size_t i = (size_t)blockIdx.x * blockDim.x + threadIdx.x;
  if (i >= (size_t)E * CH) return;
  const int e = (int)(i / CH);
  const int f = (int)(i % CH) * 4;
  const float w = ew[e];
  const float4 s = *(const float4*)(S + (size_t)src[e] * D + f);
  float* base = A + (size_t)dst[e] * D + f;
  unsafeAtomicAdd(base + 0, w * s.x);
  unsafeAtomicAdd(base + 1, w * s.y);
  unsafeAtomicAdd(base + 2, w * s.z);
  unsafeAtomicAdd(base + 3, w * s.w);
}

// Scalar variant for D = 10 (final layer); D compile-time -> magic-multiply.
template <int D>
__global__ __launch_bounds__(256) void agg_scalar(
    const float* __restrict__ S, const int* __restrict__ src,
    const int* __restrict__ dst, const float* __restrict__ ew,
    float* __restrict__ A, int E) {
  size_t i = (size_t)blockIdx.x * blockDim.x + threadIdx.x;
  if (i >= (size_t)E * D) return;
  const int e = (int)(i / D);
  const int f = (int)(i % D);
  const float w = ew[e];
  const float s = S[(size_t)src[e] * D + f];
  unsafeAtomicAdd(A + (size_t)dst[e] * D + f, w * s);
}

// ---------------------------------------------------------------------------
// Host-side orchestration (graph-capture safe: launches only, on `stream`).
// ---------------------------------------------------------------------------
static inline int cdiv(long long a, long long b) { return (int)((a + b - 1) / b); }

extern "C" void kernel_launch(void* const* d_in, const int* in_sizes, int n_in,
                              void* d_out, int out_size, void* d_ws, size_t ws_size,
                              hipStream_t stream) {
  const float* x   = (const float*)d_in[0];   // [N, 256]
  const int*   src = (const int*)  d_in[1];   // [E]
  const int*   dst = (const int*)  d_in[2];   // [E]
  const float* ew  = (const float*)d_in[3];   // [E]
  const float* W1  = (const float*)d_in[4];   // [256, 128]
  const float* b1  = (const float*)d_in[5];   // [128]
  const float* W2  = (const float*)d_in[6];   // [128, 64]
  const float* b2  = (const float*)d_in[7];   // [64]
  const float* W3  = (const float*)d_in[8];   // [64, 10]
  const float* b3  = (const float*)d_in[9];   // [10]
  float* out = (float*)d_out;                 // [N, 10]

  const int N = N_NODES, E = N_EDGES;
  const int rowTiles = N / 16;                // 3125 (N is a multiple of 16)

  // Workspace: two ping-pong buffers of N*128 floats (51.2 MB total).
  float* bufS = (float*)d_ws;                 // supports (S1 -> S2 -> S3)
  float* bufH = bufS + (size_t)N * 128;       // activations (H1 -> H2, pre-ReLU)

  // -------- Layer 1: 256 -> 128 --------
  gemm_wmma_f32<256, 128, false><<<rowTiles, 256, 0, stream>>>(x, W1, bufS, N);
  init_bias<128><<<cdiv((long long)N * 128, 256), 256, 0, stream>>>(bufH, b1, N);
  agg_vec4<128><<<cdiv((long long)E * 32, 256), 256, 0, stream>>>(
      bufS, src, dst, ew, bufH, E);
  // ReLU(H1) fused into the next GEMM's A-loads.

  // -------- Layer 2: 128 -> 64 --------
  gemm_wmma_f32<128, 64, true><<<rowTiles, 128, 0, stream>>>(bufH, W2, bufS, N);
  init_bias<64><<<cdiv((long long)N * 64, 256), 256, 0, stream>>>(bufH, b2, N);
  agg_vec4<64><<<cdiv((long long)E * 16, 256), 256, 0, stream>>>(
      bufS, src, dst, ew, bufH, E);
  // ReLU(H2) fused into the next GEMM's A-loads.

  // -------- Layer 3: 64 -> 10 (no ReLU on output) --------
  gemm_wmma_f32<64, 10, true><<<rowTiles, 32, 0, stream>>>(bufH, W3, bufS, N);
  init_bias<10><<<cdiv((long long)N * 10, 256), 256, 0, stream>>>(out, b3, N);
  agg_scalar<10><<<cdiv((long long)E * 10, 256), 256, 0, stream>>>(
      bufS, src, dst, ew, out, E);
}